// SingleGATModel_33887291965956
// MI455X (gfx1250) — compile-verified
//
#include <hip/hip_runtime.h>
#include <hip/hip_bf16.h>
#include <math.h>

#define NNODES 65536
#define NFEAT  256
#define HIDD   128
#define NGRAPH 256
#define NPGC   256
#define KTOP   205
#define NEGS   0.2f
#define LNEPS  1e-5f

typedef __attribute__((ext_vector_type(2))) float v2f;
typedef __attribute__((ext_vector_type(8))) float v8f;
typedef __attribute__((ext_vector_type(4))) int   v4i;

#define LDS_PAD 4   // floats of row padding to spread LDS banks

// 16-byte global -> LDS copy; async (ASYNCcnt) when the toolchain exposes it.
__device__ __forceinline__ void copy16_g2lds(float* lds_dst, const float* gsrc)
{
#if __has_builtin(__builtin_amdgcn_global_load_async_to_lds_b128)
    typedef __attribute__((address_space(1))) v4i* gp_t;
    typedef __attribute__((address_space(3))) v4i* lp_t;
    __builtin_amdgcn_global_load_async_to_lds_b128(
        (gp_t)(v4i*)gsrc, (lp_t)(v4i*)lds_dst, 0, 0);
#else
    *(float4*)lds_dst = *(const float4*)gsrc;
#endif
}

__device__ __forceinline__ void async_copy_fence()
{
#if __has_builtin(__builtin_amdgcn_global_load_async_to_lds_b128)
#if __has_builtin(__builtin_amdgcn_s_wait_asynccnt)
    __builtin_amdgcn_s_wait_asynccnt(0);
#else
    asm volatile("s_wait_asynccnt 0x0" ::: "memory");
#endif
#endif
    __syncthreads();
}

// ---------------------------------------------------------------------------
// WMMA f32 GEMM:  out[M,N] = epi( A[M,K] @ W[N,K]^T + bias )
// Block = 256 threads (8 waves). Each block owns a 16-row strip of A, staged
// once into LDS (async-to-LDS when available) and shared by all 8 waves; wave
// w computes column tile (blockIdx.y*8 + w). W fragments stream from global
// (L2-resident). EPI: 0 = raw store, 1 = +bias, BN(eval), ReLU.
// A-fragment (16x4 f32): lanes 0-15 -> K=k,k+1 ; lanes 16-31 -> K=k+2,k+3.
// C/D: VGPR j -> row j (lanes 0-15) / row j+8 (lanes 16-31).
// ---------------------------------------------------------------------------
template<int EPI, int KDIM>
__global__ __launch_bounds__(256)
void wmma_gemm_kernel(const float* __restrict__ A,
                      const float* __restrict__ W,
                      const float* __restrict__ bias,
                      const float* __restrict__ bnw,
                      const float* __restrict__ bnb,
                      const float* __restrict__ rm,
                      const float* __restrict__ rv,
                      float* __restrict__ out,
                      int N)
{
    __shared__ float As[16 * (KDIM + LDS_PAD)];

    const int tid    = threadIdx.x;
    const int wave   = tid >> 5;
    const int lane   = tid & 31;
    const int tile_m = blockIdx.x;
    const int tile_n = blockIdx.y * 8 + wave;

    // ---- stage 16 x KDIM A-strip into LDS (16B chunks, padded rows) ----
    const float* __restrict__ Abase = A + (size_t)tile_m * 16 * KDIM;
    const int nchunk = 16 * (KDIM / 4);           // 16B chunks in strip
    for (int cid = tid; cid < nchunk; cid += 256) {
        int r  = cid / (KDIM / 4);
        int c4 = cid % (KDIM / 4);
        copy16_g2lds(&As[r * (KDIM + LDS_PAD) + c4 * 4],
                     Abase + (size_t)r * KDIM + c4 * 4);
    }
    async_copy_fence();

    const int half = lane >> 4;    // 0 or 1
    const int l16  = lane & 15;

    const float* __restrict__ arow = &As[l16 * (KDIM + LDS_PAD) + half * 2];
    const int col = tile_n * 16 + l16;
    const float* __restrict__ wrow = W + (size_t)col * KDIM + half * 2;

    v8f acc = {};
    for (int k = 0; k < KDIM; k += 4) {
        v2f a; a.x = arow[k]; a.y = arow[k + 1];
        v2f b; b.x = wrow[k]; b.y = wrow[k + 1];
        acc = __builtin_amdgcn_wmma_f32_16x16x4_f32(
            false, a, false, b, (short)0, acc, false, false);
    }

    float bi = bias ? bias[col] : 0.0f;
    float s = 1.0f, sh = 0.0f, mu = 0.0f;
    if (EPI == 1) {
        mu = rm[col];
        s  = rsqrtf(rv[col] + LNEPS) * bnw[col];
        sh = bnb[col];
    }
    #pragma unroll
    for (int j = 0; j < 8; ++j) {
        int m = tile_m * 16 + half * 8 + j;
        float v = acc[j] + bi;
        if (EPI == 1) {
            v = (v - mu) * s + sh;
            v = fmaxf(v, 0.0f);
        }
        out[(size_t)m * N + col] = v;
    }
}

// ---------------------------------------------------------------------------
// Per-node attention coefficients: a_s[n,h] = sum_c ht[n,h*C+c] * a_src[h,c]
// ---------------------------------------------------------------------------
__global__ void attn_coef_kernel(const float* __restrict__ ht,
                                 const float* __restrict__ a_src,
                                 const float* __restrict__ a_dst,
                                 float* __restrict__ as_out,
                                 float* __restrict__ ad_out,
                                 int H, int C)
{
    int n = blockIdx.x * blockDim.x + threadIdx.x;
    if (n >= NNODES) return;
    const float* hp = ht + (size_t)n * HIDD;
    for (int h = 0; h < H; ++h) {
        float s = 0.0f, d = 0.0f;
        for (int c = 0; c < C; ++c) {
            float v = hp[h * C + c];
            s += v * a_src[h * C + c];
            d += v * a_dst[h * C + c];
        }
        as_out[n * H + h] = s;
        ad_out[n * H + h] = d;
    }
}

__global__ void fill_kernel(float* __restrict__ p, float v, size_t n)
{
    size_t i = (size_t)blockIdx.x * blockDim.x + threadIdx.x;
    if (i < n) p[i] = v;
}

__device__ __forceinline__ void atomicMaxFloat(float* addr, float val)
{
    if (val >= 0.0f)
        atomicMax((int*)addr, __float_as_int(val));
    else
        atomicMin((unsigned int*)addr, __float_as_uint(val));
}

// edges e < E come from edge_index; e >= E are self loops (src=dst=e-E)
__global__ void edge_max_kernel(const int* __restrict__ src,
                                const int* __restrict__ dst,
                                const float* __restrict__ as_,
                                const float* __restrict__ ad_,
                                float* __restrict__ emax,
                                int E, int H)
{
    int e = blockIdx.x * blockDim.x + threadIdx.x;
    if (e >= E + NNODES) return;
    int s_, d_;
    if (e < E) { s_ = src[e]; d_ = dst[e]; } else { s_ = d_ = e - E; }
    for (int h = 0; h < H; ++h) {
        float v = as_[s_ * H + h] + ad_[d_ * H + h];
        v = v > 0.0f ? v : NEGS * v;
        atomicMaxFloat(&emax[d_ * H + h], v);
    }
}

__global__ void edge_den_kernel(const int* __restrict__ src,
                                const int* __restrict__ dst,
                                const float* __restrict__ as_,
                                const float* __restrict__ ad_,
                                const float* __restrict__ emax,
                                float* __restrict__ denom,
                                int E, int H)
{
    int e = blockIdx.x * blockDim.x + threadIdx.x;
    if (e >= E + NNODES) return;
    int s_, d_;
    if (e < E) { s_ = src[e]; d_ = dst[e]; } else { s_ = d_ = e - E; }
    for (int h = 0; h < H; ++h) {
        float v = as_[s_ * H + h] + ad_[d_ * H + h];
        v = v > 0.0f ? v : NEGS * v;
        atomicAdd(&denom[d_ * H + h], __expf(v - emax[d_ * H + h]));
    }
}

// one wave per edge; lane handles 4 consecutive channels (128 total)
__global__ void edge_agg_kernel(const int* __restrict__ src,
                                const int* __restrict__ dst,
                                const float* __restrict__ as_,
                                const float* __restrict__ ad_,
                                const float* __restrict__ emax,
                                const float* __restrict__ denom,
                                const float* __restrict__ ht,
                                float* __restrict__ acc,
                                int E, int H, int C)
{
    int e    = blockIdx.x * (blockDim.x >> 5) + (threadIdx.x >> 5);
    int lane = threadIdx.x & 31;
    if (e >= E + NNODES) return;
    int s_, d_;
    if (e < E) { s_ = src[e]; d_ = dst[e]; } else { s_ = d_ = e - E; }
    int c0 = lane * 4;
    int h  = c0 / C;           // 4 channels never straddle a head (C=32 or 128)
    float v = as_[s_ * H + h] + ad_[d_ * H + h];
    v = v > 0.0f ? v : NEGS * v;
    float alpha = __expf(v - emax[d_ * H + h]) / denom[d_ * H + h];
    const float4 hv = *(const float4*)(ht + (size_t)s_ * HIDD + c0);
    float* ap = acc + (size_t)d_ * HIDD + c0;
    atomicAdd(ap + 0, alpha * hv.x);
    atomicAdd(ap + 1, alpha * hv.y);
    atomicAdd(ap + 2, alpha * hv.z);
    atomicAdd(ap + 3, alpha * hv.w);
}

// ---------------------------------------------------------------------------
// PyG LayerNorm(mode='graph') + bias + ReLU. One block (256 thr) per graph.
// Biased variance over all nodes*channels of the graph.
// ---------------------------------------------------------------------------
__global__ void graph_ln_kernel(const float* __restrict__ acc,
                                const float* __restrict__ bias,
                                const float* __restrict__ lnw,
                                const float* __restrict__ lnb,
                                float* __restrict__ out)
{
    int g = blockIdx.x;
    int t = threadIdx.x;     // 256 threads, one node each
    const float* np = acc + ((size_t)g * NPGC + t) * HIDD;

    float s = 0.0f, ss = 0.0f;
    for (int c = 0; c < HIDD; ++c) {
        float v = np[c] + bias[c];
        s += v; ss += v * v;
    }
    for (int o = 16; o > 0; o >>= 1) {
        s  += __shfl_down(s,  o, 32);
        ss += __shfl_down(ss, o, 32);
    }
    __shared__ float rs[8], rss[8], stats[2];
    if ((t & 31) == 0) { rs[t >> 5] = s; rss[t >> 5] = ss; }
    __syncthreads();
    if (t == 0) {
        float a = 0.0f, b = 0.0f;
        for (int i = 0; i < 8; ++i) { a += rs[i]; b += rss[i]; }
        float mean = a / (float)(NPGC * HIDD);
        float var  = b / (float)(NPGC * HIDD) - mean * mean;
        stats[0] = mean;
        stats[1] = rsqrtf(var + LNEPS);
    }
    __syncthreads();
    float mean = stats[0], rstd = stats[1];
    float* op = out + ((size_t)g * NPGC + t) * HIDD;
    for (int c = 0; c < HIDD; ++c) {
        float v = np[c] + bias[c];
        v = (v - mean) * rstd * lnw[c] + lnb[c];
        op[c] = fmaxf(v, 0.0f);
    }
}

// ---------------------------------------------------------------------------
// Pooling: max/mean/sum/topk per graph. One block (256 thr) per graph.
// gf[g] = [max(128) | mean(128) | sum(128) | topk(128)]
// ---------------------------------------------------------------------------
__global__ void pool_kernel(const float* __restrict__ feat,
                            const float* __restrict__ pvec,
                            float* __restrict__ gf)
{
    int g = blockIdx.x;
    int t = threadIdx.x;     // 256
    __shared__ float score[NPGC];
    __shared__ float w[NPGC];
    __shared__ float r8[8];
    __shared__ float inv_norm_s;

    // ||p||
    float pp = (t < HIDD) ? pvec[t] * pvec[t] : 0.0f;
    for (int o = 16; o > 0; o >>= 1) pp += __shfl_down(pp, o, 32);
    if ((t & 31) == 0) r8[t >> 5] = pp;
    __syncthreads();
    if (t == 0) {
        float pn = 0.0f;
        for (int i = 0; i < 8; ++i) pn += r8[i];
        inv_norm_s = rsqrtf(pn);
    }
    __syncthreads();
    float inv_norm = inv_norm_s;

    const float* x = feat + (size_t)g * NPGC * HIDD;
    const float* xp = x + (size_t)t * HIDD;
    float d = 0.0f;
    for (int c = 0; c < HIDD; ++c) d += xp[c] * pvec[c];
    score[t] = tanhf(d * inv_norm);
    __syncthreads();

    // rank among graph (top_k tie-break: lower index first)
    float mys = score[t];
    int rank = 0;
    for (int i = 0; i < NPGC; ++i) {
        float si = score[i];
        rank += (si > mys) || (si == mys && i < t);
    }
    w[t] = (rank < KTOP) ? mys : 0.0f;
    __syncthreads();

    if (t < HIDD) {
        float mx = -INFINITY, sm = 0.0f, tk = 0.0f;
        for (int i = 0; i < NPGC; ++i) {
            float v = x[(size_t)i * HIDD + t];
            mx = fmaxf(mx, v);
            sm += v;
            tk += w[i] * v;
        }
        float* o = gf + (size_t)g * (4 * HIDD);
        o[t]            = mx;
        o[HIDD + t]     = sm / (float)NPGC;
        o[2 * HIDD + t] = sm;
        o[3 * HIDD + t] = tk / (float)KTOP;
    }
}

__global__ void cls_final_kernel(const float* __restrict__ h2,
                                 const float* __restrict__ W3,
                                 const float* __restrict__ b3,
                                 float* __restrict__ out)
{
    int i = blockIdx.x * blockDim.x + threadIdx.x;  // G*2
    if (i >= NGRAPH * 2) return;
    int g = i >> 1, o = i & 1;
    float s = b3[o];
    const float* hp = h2 + (size_t)g * HIDD;
    const float* wp = W3 + (size_t)o * HIDD;
    for (int c = 0; c < HIDD; ++c) s += hp[c] * wp[c];
    out[i] = s;
}

// ---------------------------------------------------------------------------

static inline unsigned cdiv(size_t a, unsigned b) { return (unsigned)((a + b - 1) / b); }

extern "C" void kernel_launch(void* const* d_in, const int* in_sizes, int n_in,
                              void* d_out, int out_size, void* d_ws, size_t ws_size,
                              hipStream_t stream)
{
    // ---- inputs (setup_inputs insertion order) ----
    const float* x    = (const float*)d_in[0];
    const int*   eidx = (const int*)d_in[1];
    const int E = in_sizes[1] / 2;
    const int* esrc = eidx;
    const int* edst = eidx + E;
    // d_in[2] = batch (unused; graphs are equal-sized & sorted)
    const float* W_in  = (const float*)d_in[3];
    const float* b_in  = (const float*)d_in[4];
    const float* bni_w = (const float*)d_in[5];
    const float* bni_b = (const float*)d_in[6];
    const float* bni_m = (const float*)d_in[7];
    const float* bni_v = (const float*)d_in[8];

    struct Gat { const float *W, *a_src, *a_dst, *bias, *lnw, *lnb; int H, C; };
    Gat gat[3];
    int bi = 9;
    for (int l = 0; l < 3; ++l) {
        gat[l].W     = (const float*)d_in[bi + 0];
        gat[l].a_src = (const float*)d_in[bi + 1];
        gat[l].a_dst = (const float*)d_in[bi + 2];
        gat[l].bias  = (const float*)d_in[bi + 3];
        gat[l].lnw   = (const float*)d_in[bi + 4];
        gat[l].lnb   = (const float*)d_in[bi + 5];
        bi += 6;
    }
    gat[0].H = 4; gat[0].C = 32;
    gat[1].H = 4; gat[1].C = 32;
    gat[2].H = 1; gat[2].C = 128;

    const float* topk_p = (const float*)d_in[27];
    const float* c_W1 = (const float*)d_in[28];
    const float* c_b1 = (const float*)d_in[29];
    const float* c_bn1w = (const float*)d_in[30];
    const float* c_bn1b = (const float*)d_in[31];
    const float* c_bn1m = (const float*)d_in[32];
    const float* c_bn1v = (const float*)d_in[33];
    const float* c_W2 = (const float*)d_in[34];
    const float* c_b2 = (const float*)d_in[35];
    const float* c_bn2w = (const float*)d_in[36];
    const float* c_bn2b = (const float*)d_in[37];
    const float* c_bn2m = (const float*)d_in[38];
    const float* c_bn2v = (const float*)d_in[39];
    const float* c_W3 = (const float*)d_in[40];
    const float* c_b3 = (const float*)d_in[41];

    // ---- workspace layout (floats) ----
    float* ws = (float*)d_ws;
    const size_t NF = (size_t)NNODES * HIDD;
    float* h     = ws;
    float* ht    = h + NF;
    float* acc   = ht + NF;
    float* as_   = acc + NF;
    float* ad_   = as_ + (size_t)NNODES * 4;
    float* emax  = ad_ + (size_t)NNODES * 4;
    float* denom = emax + (size_t)NNODES * 4;
    float* gf    = denom + (size_t)NNODES * 4;
    float* h1    = gf + (size_t)NGRAPH * 4 * HIDD;
    float* h2    = h1 + (size_t)NGRAPH * 2 * HIDD;

    float* logits   = (float*)d_out;
    float* features = logits + NGRAPH * 2;

    const int ET = E + NNODES;   // edges + self loops

    // 1) input projection: h = relu(bn(x @ W_in^T + b_in))
    {
        dim3 grid(NNODES / 16, HIDD / 128);
        wmma_gemm_kernel<1, NFEAT><<<grid, 256, 0, stream>>>(
            x, W_in, b_in, bni_w, bni_b, bni_m, bni_v, h, HIDD);
    }

    // 2) three GAT layers
    for (int l = 0; l < 3; ++l) {
        const Gat& g = gat[l];
        float* lout = (l == 2) ? features : h;

        // 2a) feature transform ht = h @ W^T
        {
            dim3 grid(NNODES / 16, HIDD / 128);
            wmma_gemm_kernel<0, HIDD><<<grid, 256, 0, stream>>>(
                h, g.W, nullptr, nullptr, nullptr, nullptr, nullptr,
                ht, HIDD);
        }
        // 2b) per-node attention logits
        attn_coef_kernel<<<cdiv(NNODES, 256), 256, 0, stream>>>(
            ht, g.a_src, g.a_dst, as_, ad_, g.H, g.C);

        // 2c) init reductions
        fill_kernel<<<cdiv((size_t)NNODES * g.H, 256), 256, 0, stream>>>(
            emax, -INFINITY, (size_t)NNODES * g.H);
        fill_kernel<<<cdiv((size_t)NNODES * g.H, 256), 256, 0, stream>>>(
            denom, 0.0f, (size_t)NNODES * g.H);
        fill_kernel<<<cdiv(NF, 256), 256, 0, stream>>>(acc, 0.0f, NF);

        // 2d) edge softmax + aggregation
        edge_max_kernel<<<cdiv(ET, 256), 256, 0, stream>>>(
            esrc, edst, as_, ad_, emax, E, g.H);
        edge_den_kernel<<<cdiv(ET, 256), 256, 0, stream>>>(
            esrc, edst, as_, ad_, emax, denom, E, g.H);
        edge_agg_kernel<<<cdiv(ET, 8), 256, 0, stream>>>(
            esrc, edst, as_, ad_, emax, denom, ht, acc, E, g.H, g.C);

        // 2e) bias + graph layernorm + relu
        graph_ln_kernel<<<NGRAPH, NPGC, 0, stream>>>(
            acc, g.bias, g.lnw, g.lnb, lout);
    }

    // 3) multi-scale pooling -> gf [G, 512]
    pool_kernel<<<NGRAPH, NPGC, 0, stream>>>(features, topk_p, gf);

    // 4) classifier
    {
        dim3 grid(NGRAPH / 16, (2 * HIDD) / 128);
        wmma_gemm_kernel<1, 4 * HIDD><<<grid, 256, 0, stream>>>(
            gf, c_W1, c_b1, c_bn1w, c_bn1b, c_bn1m, c_bn1v,
            h1, 2 * HIDD);
    }
    {
        dim3 grid(NGRAPH / 16, HIDD / 128);
        wmma_gemm_kernel<1, 2 * HIDD><<<grid, 256, 0, stream>>>(
            h1, c_W2, c_b2, c_bn2w, c_bn2b, c_bn2m, c_bn2v,
            h2, HIDD);
    }
    cls_final_kernel<<<cdiv(NGRAPH * 2, 256), 256, 0, stream>>>(
        h2, c_W3, c_b3, logits);
}